// local_loss_idx_40596030882289
// MI455X (gfx1250) — compile-verified
//
#include <hip/hip_runtime.h>
#include <math.h>

typedef __attribute__((ext_vector_type(2))) float v2f;
typedef __attribute__((ext_vector_type(8))) float v8f;

#define NS   8192
#define MM   16
#define DD   512
#define WPB  8   // waves per block (wave32)

// One wave (32 lanes) handles one sample: two 16x16x512 Gram matrices via
// V_WMMA_F32_16X16X4_F32, norms folded in, tanh-distance, then a 31-step
// anti-diagonal DTW (ap in lanes 0-15, an in lanes 16-31).
__global__ __launch_bounds__(256)
void local_dtw_kernel(const float* __restrict__ lf,
                      const int*   __restrict__ p_inds,
                      const int*   __restrict__ n_inds,
                      float* __restrict__ dist_ap,
                      float* __restrict__ dist_an) {
    __shared__ float tiles[WPB][2][MM * MM];

    const int lane = threadIdx.x & 31;
    const int wave = threadIdx.x >> 5;
    const int s    = blockIdx.x * WPB + wave;

    const float* x  = lf + (size_t)s * (MM * DD);
    const float* yp = lf + (size_t)p_inds[s] * (MM * DD);
    const float* yn = lf + (size_t)n_inds[s] * (MM * DD);

    // A operand: lane -> row M=lane&15, K-half (lane>>4)*2 (+0,+1)
    // B operand: lane -> col N=lane&15, same K-half (B[k][n] = y[n][k])
    // => identical float2 address pattern for A and B sources.
    const int    rc   = lane & 15;
    const int    kh   = (lane >> 4) << 1;
    const size_t base = (size_t)rc * DD + kh;

    v8f cp = {}; v8f cn = {};
    v2f sxv = {0.f, 0.f}, spv = {0.f, 0.f}, snv = {0.f, 0.f};

    #pragma unroll 4
    for (int k = 0; k < DD; k += 4) {
        v2f a  = *(const v2f*)(x  + base + k);
        v2f bp = *(const v2f*)(yp + base + k);
        v2f bn = *(const v2f*)(yn + base + k);
        // packed accumulators -> v_pk_fma_f32, no lane re-packing movs
        sxv += a  * a;
        spv += bp * bp;
        snv += bn * bn;
        // D = A(16x4) * B(4x16) + C  (fp32 WMMA, wave32)
        cp = __builtin_amdgcn_wmma_f32_16x16x4_f32(false, a, false, bp,
                                                   (short)0, cp, false, false);
        cn = __builtin_amdgcn_wmma_f32_16x16x4_f32(false, a, false, bn,
                                                   (short)0, cn, false, false);
    }

    float sx = sxv.x + sxv.y;
    float sp = spv.x + spv.y;
    float sn = snv.x + snv.y;

    // Lanes l and l^16 hold complementary K-halves of row (l&15) norms.
    sx += __shfl_xor(sx, 16, 32);
    sp += __shfl_xor(sp, 16, 32);
    sn += __shfl_xor(sn, 16, 32);

    const float EPS = 1e-12f;
    float nx  = sqrtf(sx) + EPS;       // normalize divisor, row lane&15 of x
    float np  = sqrtf(sp) + EPS;
    float nn  = sqrtf(sn) + EPS;
    float xn2 = sx / (nx * nx);        // ||x_norm[row]||^2 (== ~1)
    float pn2 = sp / (np * np);
    float nn2 = sn / (nn * nn);

    // C/D layout: VGPR r, lanes<16 -> M=r, lanes>=16 -> M=r+8; N = lane&15.
    float* tap = tiles[wave][0];
    float* tan_ = tiles[wave][1];
    const int mhalf = (lane >> 4) << 3;
    #pragma unroll
    for (int r = 0; r < 8; ++r) {
        int   mrow = r + mhalf;
        float nxm  = __shfl(nx,  mrow, 32);
        float xn2m = __shfl(xn2, mrow, 32);

        float g  = cp[r] / (nxm * np);           // normalized dot
        float d2 = xn2m + pn2 - 2.f * g;
        float dd = sqrtf(fmaxf(d2, EPS));
        float e  = __expf(dd);
        tap[mrow * 16 + rc] = (e - 1.f) / (e + 1.f);

        g  = cn[r] / (nxm * nn);
        d2 = xn2m + nn2 - 2.f * g;
        dd = sqrtf(fmaxf(d2, EPS));
        e  = __expf(dd);
        tan_[mrow * 16 + rc] = (e - 1.f) / (e + 1.f);
    }
    __syncthreads();

    // DTW shortest path, anti-diagonal wavefront.
    // dp[i][j] = min(dp[i-1][j], dp[i][j-1]) + t[i][j]; boundaries via +INF.
    const float* tmat = (lane < 16) ? tap : tan_;
    const int   j   = rc;
    const float INF = 3.0e38f;
    float dp = 0.f;
    for (int dgl = 0; dgl < 31; ++dgl) {
        float left = __shfl_up(dp, 1, 32);   // (i, j-1) from prev diagonal
        int i = dgl - j;
        if (i >= 0 && i < 16) {
            float t = tmat[i * 16 + j];
            float u = (i == 0) ? INF : dp;   // (i-1, j)
            float l = (j == 0) ? INF : left;
            float m = fminf(u, l);
            dp = (i == 0 && j == 0) ? t : (m + t);
        }
    }
    if (lane == 15) dist_ap[s] = dp;
    if (lane == 31) dist_an[s] = dp;
}

// Deterministic single-block tree reduction for the margin-ranking loss.
__global__ __launch_bounds__(256)
void loss_kernel(const float* __restrict__ dist_ap,
                 const float* __restrict__ dist_an,
                 float* __restrict__ out) {
    __shared__ float red[256];
    float acc = 0.f;
    for (int i = threadIdx.x; i < NS; i += 256) {
        float v = 0.3f - (dist_an[i] - dist_ap[i]);
        acc += fmaxf(v, 0.f);
    }
    red[threadIdx.x] = acc;
    __syncthreads();
    for (int s = 128; s > 0; s >>= 1) {
        if ((int)threadIdx.x < s) red[threadIdx.x] += red[threadIdx.x + s];
        __syncthreads();
    }
    if (threadIdx.x == 0) out[0] = red[0] / (float)NS;
}

extern "C" void kernel_launch(void* const* d_in, const int* in_sizes, int n_in,
                              void* d_out, int out_size, void* d_ws, size_t ws_size,
                              hipStream_t stream) {
    const float* lf = (const float*)d_in[0];
    const int*   p  = (const int*)d_in[1];
    const int*   n  = (const int*)d_in[2];
    // d_in[3] (labels) unused by the reference.
    float* out = (float*)d_out;
    float* ap  = out + 1;
    float* an  = out + 1 + NS;

    local_dtw_kernel<<<NS / WPB, 256, 0, stream>>>(lf, p, n, ap, an);
    loss_kernel<<<1, 256, 0, stream>>>(ap, an, out);
}